// activation_module_15307263442966
// MI455X (gfx1250) — compile-verified
//
#include <hip/hip_runtime.h>

#define SIZE 4096

// Clang native vector type: accepted by __builtin_nontemporal_load/store
// and lowers to global_load_b128 / global_store_b128 with th:NT.
typedef float v4f __attribute__((ext_vector_type(4)));

// out[:, 0:4096]     = p  (verbatim copy of x[:, 0:4096])
// out[:, 4096:8192]  = w[row] * selu(p) + q
// selu(x) = scale * (x > 0 ? x : alpha * expm1(x))
__global__ __launch_bounds__(256) void selu_gate_concat_kernel(
    const float* __restrict__ x,
    const float* __restrict__ weight,
    float* __restrict__ out)
{
    constexpr float kScale = 1.0507009873554804934193349852946f;
    constexpr float kAlpha = 1.6732632423543772848170429916717f;

    // 2 blocks per row; each block covers 2048 columns of the p-half
    // (2 chunks of 256 threads * float4 = 1024 columns each).
    const int row  = (int)(blockIdx.x >> 1);
    const int half = (int)(blockIdx.x & 1);

    const size_t base = (size_t)row * (size_t)(2 * SIZE);

    // Uniform per block -> scalar (SMEM) load.
    const float w = weight[row];

    const int col0 = half * 2048 + ((int)threadIdx.x << 2);

#pragma unroll
    for (int c = 0; c < 2; ++c) {
        const int col = col0 + c * 1024;

        const v4f* pPtr = (const v4f*)(x + base + col);
        const v4f* qPtr = (const v4f*)(x + base + SIZE + col);

        // Streamed once; dataset (256MB) > L2 (192MB) -> non-temporal.
        v4f p = __builtin_nontemporal_load(pPtr);
        v4f q = __builtin_nontemporal_load(qPtr);

        v4f r;
#pragma unroll
        for (int i = 0; i < 4; ++i) {
            float pi = p[i];
            float s  = kScale * (pi > 0.0f ? pi : kAlpha * expm1f(pi));
            r[i]     = fmaf(w, s, q[i]);
        }

        __builtin_nontemporal_store(p, (v4f*)(out + base + col));
        __builtin_nontemporal_store(r, (v4f*)(out + base + SIZE + col));
    }
}

extern "C" void kernel_launch(void* const* d_in, const int* in_sizes, int n_in,
                              void* d_out, int out_size, void* d_ws, size_t ws_size,
                              hipStream_t stream) {
    const float* x      = (const float*)d_in[0];   // (4096, 8192) fp32
    const float* weight = (const float*)d_in[1];   // (4096,) fp32
    float* out          = (float*)d_out;           // (4096, 8192) fp32

    dim3 grid(SIZE * 2);   // 8192 blocks: 2 per row
    dim3 block(256);       // 8 waves (wave32) per block
    hipLaunchKernelGGL(selu_gate_concat_kernel, grid, block, 0, stream,
                       x, weight, out);
}